// GLadderModule_40578851013020
// MI455X (gfx1250) — compile-verified
//
#include <hip/hip_runtime.h>
#include <hip/hip_bf16.h>

#define HDIM 128
#define RDIM 16
#define SCALING 2.0f
#define LN_EPS 1e-5f
#define MTILE 32                 // rows per block (two 16-row WMMA subtiles)
#define PITCH (HDIM + 4)         // LDS pitch: (4m+ko) spans all 64 banks -> conflict-free
#define PPITCH 18                // pitch for 16-wide P tiles (18m%64 all distinct)

typedef __attribute__((ext_vector_type(2))) float v2f;
typedef __attribute__((ext_vector_type(8))) float v8f;
typedef int b128v __attribute__((vector_size(16)));   // matches builtin's param type

#define AS1 __attribute__((address_space(1)))
#define AS3 __attribute__((address_space(3)))

// ---------------------------------------------------------------------------
// CDNA5 async copy: global -> LDS, 16B per lane, tracked by ASYNCcnt.
// Falls back to a plain LDS store if the builtin is unavailable.
// ---------------------------------------------------------------------------
__device__ __forceinline__ void async_cp16(const float* g, float* l) {
#if __has_builtin(__builtin_amdgcn_global_load_async_to_lds_b128)
  __builtin_amdgcn_global_load_async_to_lds_b128(
      (AS1 b128v*)(b128v*)(void*)g, (AS3 b128v*)(b128v*)(void*)l, 0, 0);
#else
  *(float4*)l = *(const float4*)g;
#endif
}

__device__ __forceinline__ void async_wait0() {
  asm volatile("s_wait_asynccnt 0" ::: "memory");
}

// ---------------------------------------------------------------------------
// Fragment loader for V_WMMA_F32_16X16X4_F32 (ISA 7.12.2):
//   lane L: row (L&15), K-pair k + 2*(L>>4)  -> contiguous float2.
// Works for row-major A tiles and for row-major [N][K] weights used as B^T.
// ---------------------------------------------------------------------------
__device__ __forceinline__ v2f frag_ld(const float* __restrict__ p,
                                       int rowbase, int stride, int k, int lane) {
  int m  = lane & 15;
  int ko = k + ((lane >> 4) << 1);
  const float* q = p + (size_t)(rowbase + m) * stride + ko;
  v2f r; r.x = q[0]; r.y = q[1];
  return r;
}

#define WMMA_F32(a, b, c) \
  __builtin_amdgcn_wmma_f32_16x16x4_f32(false, (a), false, (b), (short)0, (c), false, false)

// ---- message-passing helper kernels ---------------------------------------

__global__ void k_init(float* __restrict__ deg, float* __restrict__ summed,
                       int n_nodes, int total) {
  int i = blockIdx.x * blockDim.x + threadIdx.x;
  if (i < total) summed[i] = 0.0f;
  if (i < n_nodes) deg[i] = 1.0f;          // self loop contributes 1
}

__global__ void k_degree(const long long* __restrict__ dst,
                         float* __restrict__ deg, int n_edges) {
  int e = blockIdx.x * blockDim.x + threadIdx.x;
  if (e < n_edges) atomicAdd(&deg[(int)dst[e]], 1.0f);
}

__global__ void k_dinv(const float* __restrict__ deg, float* __restrict__ dinv,
                       int n_nodes) {
  int i = blockIdx.x * blockDim.x + threadIdx.x;
  if (i < n_nodes) dinv[i] = 1.0f / sqrtf(deg[i]);   // deg >= 1 always
}

// one wave (32 lanes) per edge; lane handles 4 contiguous channels
__global__ __launch_bounds__(256) void k_scatter(
    const float* __restrict__ x, const long long* __restrict__ src,
    const long long* __restrict__ dst, const float* __restrict__ dinv,
    float* __restrict__ summed, int n_edges) {
  int wave = (blockIdx.x * blockDim.x + threadIdx.x) >> 5;
  int lane = threadIdx.x & 31;
  if (wave >= n_edges) return;
  long long r = src[wave];
  long long c = dst[wave];
  float norm = dinv[r] * dinv[c];
  float4 v = *(const float4*)(x + (size_t)r * HDIM + lane * 4);
  float* s = summed + (size_t)c * HDIM + lane * 4;
  atomicAdd(s + 0, norm * v.x);
  atomicAdd(s + 1, norm * v.y);
  atomicAdd(s + 2, norm * v.z);
  atomicAdd(s + 3, norm * v.w);
}

// msg = (summed + dinv^2 * x) / deg   (self-loop added here, in place)
__global__ void k_msgfin(const float* __restrict__ x, const float* __restrict__ deg,
                         const float* __restrict__ dinv, float* __restrict__ summed,
                         int total) {
  int idx = blockIdx.x * blockDim.x + threadIdx.x;
  if (idx >= total) return;
  int i = idx >> 7;                                   // / HDIM
  float d = dinv[i];
  summed[idx] = (summed[idx] + d * d * x[idx]) / deg[i];
}

// ---- fused: LoRA-linear x2 + sigmoid gate + residual + LayerNorm ----------
// Block = 32 node rows (2 WMMA subtiles), 8 waves; wave w owns cols [16w,16w+16).
// Activations async-staged in LDS once per block; W fragments reused by 2 WMMAs.

__global__ __launch_bounds__(256) void k_fused(
    const float* __restrict__ hidden, const float* __restrict__ msg,
    const float* __restrict__ msg_W, const float* __restrict__ msg_b,
    const float* __restrict__ msg_A, const float* __restrict__ msg_B,
    const float* __restrict__ gate_W, const float* __restrict__ gate_b,
    const float* __restrict__ gate_A, const float* __restrict__ gate_B,
    const float* __restrict__ gamma, const float* __restrict__ beta,
    float* __restrict__ out) {
  __shared__ float lds_msg[MTILE * PITCH];
  __shared__ float lds_hid[MTILE * PITCH];
  __shared__ float Pm[MTILE * PPITCH];
  __shared__ float Pg[MTILE * PPITCH];

  int lane = threadIdx.x & 31;
  int w = threadIdx.x >> 5;
  int M0 = blockIdx.x * MTILE;
  int N0 = w * 16;
  int t = threadIdx.x;

  // Stage 0: async-copy both 32x128 activation tiles into LDS (16B/lane)
#pragma unroll
  for (int j = 0; j < 4; ++j) {
    int idx = t * 4 + j;                  // 1024 float4 slots per tile
    int r = idx >> 5;                     // 32 float4 per row
    int c = (idx & 31) * 4;
    async_cp16(msg    + (size_t)(M0 + r) * HDIM + c, lds_msg + r * PITCH + c);
    async_cp16(hidden + (size_t)(M0 + r) * HDIM + c, lds_hid + r * PITCH + c);
  }
  async_wait0();
  __syncthreads();

  // Stage 1: LoRA down-projections P = X @ A^T. Four independent 16-row jobs:
  // wave0: msg/rows0-15, wave1: msg/rows16-31, wave2: gate/0-15, wave3: gate/16-31
  if (w < 4) {
    const float* X = (w < 2) ? lds_msg : lds_hid;
    const float* A = (w < 2) ? msg_A : gate_A;
    int sub = (w & 1) * 16;
    v8f p = {};
    for (int k = 0; k < HDIM; k += 4) {
      v2f a = frag_ld(X, sub, PITCH, k, lane);
      v2f b = frag_ld(A, 0, HDIM, k, lane);
      p = WMMA_F32(a, b, p);
    }
    float* P = (w < 2) ? Pm : Pg;
    int col = lane & 15, rbse = (lane >> 4) * 8;
#pragma unroll
    for (int v = 0; v < 8; ++v) P[(sub + rbse + v) * PPITCH + col] = p[v];
  }
  __syncthreads();

  // Stage 2: main 128-deep GEMMs; each W/gateW fragment feeds 2 WMMAs (M reuse)
  v8f am[2] = {{}, {}}, ag[2] = {{}, {}};
  for (int k = 0; k < HDIM; k += 4) {
    v2f b0 = frag_ld(msg_W, N0, HDIM, k, lane);
    v2f b1 = frag_ld(gate_W, N0, HDIM, k, lane);
    v2f a00 = frag_ld(lds_msg, 0, PITCH, k, lane);
    v2f a01 = frag_ld(lds_msg, 16, PITCH, k, lane);
    v2f a10 = frag_ld(lds_hid, 0, PITCH, k, lane);
    v2f a11 = frag_ld(lds_hid, 16, PITCH, k, lane);
    am[0] = WMMA_F32(a00, b0, am[0]);
    am[1] = WMMA_F32(a01, b0, am[1]);
    ag[0] = WMMA_F32(a10, b1, ag[0]);
    ag[1] = WMMA_F32(a11, b1, ag[1]);
  }

  // Stage 3: LoRA up-projection (K = 16) from LDS-staged P
  v8f lm[2] = {{}, {}}, lg[2] = {{}, {}};
  for (int k = 0; k < RDIM; k += 4) {
    v2f bp = frag_ld(msg_B, N0, RDIM, k, lane);
    v2f bq = frag_ld(gate_B, N0, RDIM, k, lane);
    lm[0] = WMMA_F32(frag_ld(Pm, 0, PPITCH, k, lane), bp, lm[0]);
    lm[1] = WMMA_F32(frag_ld(Pm, 16, PPITCH, k, lane), bp, lm[1]);
    lg[0] = WMMA_F32(frag_ld(Pg, 0, PPITCH, k, lane), bq, lg[0]);
    lg[1] = WMMA_F32(frag_ld(Pg, 16, PPITCH, k, lane), bq, lg[1]);
  }

  // Stage 4: bias + LoRA scale + sigmoid gate + residual (residual from LDS)
  int n = N0 + (lane & 15);
  float bm = msg_b[n], bg = gate_b[n];
  int rbse = (lane >> 4) * 8;
  float xv[16];
#pragma unroll
  for (int sub = 0; sub < 2; ++sub)
#pragma unroll
    for (int v = 0; v < 8; ++v) {
      int m = sub * 16 + rbse + v;
      float mt = am[sub][v] + bm + SCALING * lm[sub][v];
      float gz = ag[sub][v] + bg + SCALING * lg[sub][v];
      float gate = 1.0f / (1.0f + expf(-gz));
      xv[sub * 8 + v] = lds_hid[m * PITCH + n] + gate * mt;
    }
  __syncthreads();                       // tile reads done; recycle lds_msg
  float* xt = lds_msg;
#pragma unroll
  for (int sub = 0; sub < 2; ++sub)
#pragma unroll
    for (int v = 0; v < 8; ++v) {
      int m = sub * 16 + rbse + v;
      xt[m * PITCH + n] = xv[sub * 8 + v];
    }
  __syncthreads();

  // Stage 5: LayerNorm, wave w handles rows 4w..4w+3 (shuffle reduction)
#pragma unroll
  for (int rr = 0; rr < 4; ++rr) {
    int r = 4 * w + rr;
    int gr = M0 + r;
    int c = lane * 4;
    const float4 x4 = *(const float4*)&xt[r * PITCH + c];
    float s  = x4.x + x4.y + x4.z + x4.w;
    float sq = x4.x * x4.x + x4.y * x4.y + x4.z * x4.z + x4.w * x4.w;
    for (int off = 16; off > 0; off >>= 1) {
      s  += __shfl_xor(s, off, 32);
      sq += __shfl_xor(sq, off, 32);
    }
    float mu   = s * (1.0f / HDIM);
    float var  = sq * (1.0f / HDIM) - mu * mu;
    float rstd = 1.0f / sqrtf(var + LN_EPS);
    float4 g4 = *(const float4*)&gamma[c];
    float4 b4 = *(const float4*)&beta[c];
    float4 o;
    o.x = (x4.x - mu) * rstd * g4.x + b4.x;
    o.y = (x4.y - mu) * rstd * g4.y + b4.y;
    o.z = (x4.z - mu) * rstd * g4.z + b4.z;
    o.w = (x4.w - mu) * rstd * g4.w + b4.w;
    *(float4*)&out[(size_t)gr * HDIM + c] = o;
  }
}

// ---------------------------------------------------------------------------

extern "C" void kernel_launch(void* const* d_in, const int* in_sizes, int n_in,
                              void* d_out, int out_size, void* d_ws, size_t ws_size,
                              hipStream_t stream) {
  const float*     hidden = (const float*)d_in[0];
  const long long* ei     = (const long long*)d_in[1];   // int64 [2][E]
  const float*     msg_W  = (const float*)d_in[2];
  const float*     msg_b  = (const float*)d_in[3];
  const float*     msg_A  = (const float*)d_in[4];
  const float*     msg_B  = (const float*)d_in[5];
  const float*     gate_W = (const float*)d_in[6];
  const float*     gate_b = (const float*)d_in[7];
  const float*     gate_A = (const float*)d_in[8];
  const float*     gate_B = (const float*)d_in[9];
  const float*     gamma  = (const float*)d_in[10];
  const float*     beta   = (const float*)d_in[11];
  float* out = (float*)d_out;

  int N = in_sizes[0] / HDIM;
  int E = in_sizes[1] / 2;
  int total = N * HDIM;

  float* deg    = (float*)d_ws;        // N
  float* dinv   = deg + N;             // N
  float* summed = dinv + N;            // N*H, finalized to msg in place

  const int tpb = 256;
  k_init  <<<(total + tpb - 1) / tpb, tpb, 0, stream>>>(deg, summed, N, total);
  k_degree<<<(E + tpb - 1) / tpb, tpb, 0, stream>>>(ei + E, deg, E);
  k_dinv  <<<(N + tpb - 1) / tpb, tpb, 0, stream>>>(deg, dinv, N);
  k_scatter<<<(E + 7) / 8, 256, 0, stream>>>(hidden, ei, ei + E, dinv, summed, E);
  k_msgfin<<<(total + tpb - 1) / tpb, tpb, 0, stream>>>(hidden, deg, dinv, summed, total);
  k_fused <<<N / MTILE, 256, 0, stream>>>(hidden, summed, msg_W, msg_b, msg_A, msg_B,
                                          gate_W, gate_b, gate_A, gate_B, gamma, beta, out);
}